// Encoder_66735201845341
// MI455X (gfx1250) — compile-verified
//
#include <hip/hip_runtime.h>
#include <hip/hip_bf16.h>

typedef float v2f __attribute__((ext_vector_type(2)));
typedef float v8f __attribute__((ext_vector_type(8)));

#define COLS 256
#define OUTS 128
#define MAX_BLOCKS 2048

// ---------------------------------------------------------------------------
// Kernel 1: per-block partial column sums of x [rows, 256].
// 256 threads = 4 row-groups x 64 float4 column-slots -> fully coalesced
// 1024B row segments (global_load_b128). Deterministic: fixed per-thread
// accumulation order, fixed in-block cross-group reduction order, partials
// written to distinct workspace slots (no FP atomics).
// ---------------------------------------------------------------------------
__global__ void __launch_bounds__(256)
enc_colsum_partial(const float* __restrict__ x, float* __restrict__ partial,
                   int rows, int rowsPerBlock) {
    __shared__ float sm[4][COLS];
    const int tid = threadIdx.x;
    const int g   = tid >> 6;        // row-group 0..3
    const int c4  = tid & 63;        // float4 column slot 0..63

    long r0 = (long)blockIdx.x * rowsPerBlock;
    long r1 = r0 + rowsPerBlock;
    if (r1 > rows) r1 = rows;

    float ax = 0.f, ay = 0.f, az = 0.f, aw = 0.f;
    for (long r = r0 + g; r < r1; r += 4) {
        const float4 v = ((const float4*)(x + (size_t)r * COLS))[c4];
        ax += v.x; ay += v.y; az += v.z; aw += v.w;
    }
    sm[g][c4 * 4 + 0] = ax;
    sm[g][c4 * 4 + 1] = ay;
    sm[g][c4 * 4 + 2] = az;
    sm[g][c4 * 4 + 3] = aw;
    __syncthreads();

    // column tid: sum the 4 row-groups in fixed order
    float s = sm[0][tid] + sm[1][tid] + sm[2][tid] + sm[3][tid];
    partial[(size_t)blockIdx.x * COLS + tid] = s;
}

// ---------------------------------------------------------------------------
// Kernel 2 (single block, 256 threads = 8 waves):
//   phase 1: deterministic reduction of block partials -> colsum[256] in LDS
//            (4 strided accumulators for ILP; fixed pattern => deterministic)
//   phase 2: h = colsum @ W^T via V_WMMA_F32_16X16X4_F32.
// Formulation: D = A x B, A[16x4] has colsum chunk in row M=0 (rows 1..15
// zero via hoisted multiplicative mask -> no EXEC manipulation in the loop),
// B[4x16] = W^T tile. Wave w owns output columns [16w, 16w+16).
// A layout (32-bit 16x4): lane L -> M=L%16, VGPR v -> K=(L/16)*2+v.
// B layout (4x16):        lane L -> N=L%16, VGPR v -> K=(L/16)*2+v.
// D layout: VGPR0, lanes 0..15 -> (M=0, N=lane)  => direct per-lane store.
// ---------------------------------------------------------------------------
__global__ void __launch_bounds__(256)
enc_finish_wmma(const float* __restrict__ partial, const float* __restrict__ W,
                float* __restrict__ out, int nblocks) {
    __shared__ float cs[COLS];
    const int tid = threadIdx.x;

    // deterministic partial reduction: fixed strided pattern over blocks
    float s0 = 0.f, s1 = 0.f, s2 = 0.f, s3 = 0.f;
    int b = 0;
    for (; b + 4 <= nblocks; b += 4) {
        s0 += partial[(size_t)(b + 0) * COLS + tid];
        s1 += partial[(size_t)(b + 1) * COLS + tid];
        s2 += partial[(size_t)(b + 2) * COLS + tid];
        s3 += partial[(size_t)(b + 3) * COLS + tid];
    }
    for (; b < nblocks; ++b)
        s0 += partial[(size_t)b * COLS + tid];
    cs[tid] = (s0 + s1) + (s2 + s3);
    __syncthreads();

    const int lane  = tid & 31;
    const int wave  = tid >> 5;            // 0..7 -> N tile
    const int nbase = wave * 16;
    const int klane = (lane >> 4) << 1;    // 0 for lanes 0-15, 2 for 16-31
    const float m   = ((lane & 15) == 0) ? 1.f : 0.f;  // A-matrix row M=0 mask
    const int n     = nbase + (lane & 15); // output column / W row

    const float* wrow = W + (size_t)n * COLS + klane;

    v8f c = {};
    #pragma unroll 4
    for (int k0 = 0; k0 < COLS; k0 += 4) {
        // unconditional 8B loads (indices are even -> 8B aligned), masked by mul
        const v2f cv = *(const v2f*)&cs[k0 + klane];
        const v2f wv = *(const v2f*)&wrow[k0];
        v2f a, bb;
        a.x  = cv.x * m;
        a.y  = cv.y * m;
        bb.x = wv.x;
        bb.y = wv.y;
        // D = A*B + C ; 8-arg form: (neg_a, A, neg_b, B, c_mod, C, reuse_a, reuse_b)
        c = __builtin_amdgcn_wmma_f32_16x16x4_f32(
                false, a, false, bb, (short)0, c, false, false);
    }

    if (lane < 16)
        out[n] = c[0];   // D row M=0 lives in VGPR0 of lanes 0..15
}

extern "C" void kernel_launch(void* const* d_in, const int* in_sizes, int n_in,
                              void* d_out, int out_size, void* d_ws, size_t ws_size,
                              hipStream_t stream) {
    const float* x = (const float*)d_in[0];   // [rows, 256] f32
    const float* W = (const float*)d_in[1];   // [128, 256] f32
    float* out     = (float*)d_out;           // [128] f32
    float* partial = (float*)d_ws;

    const int rows = in_sizes[0] / COLS;

    // size grid by available workspace (deterministic for fixed ws_size)
    long maxNB = (long)(ws_size / (COLS * sizeof(float)));
    int NB = (int)(maxNB < 1 ? 1 : (maxNB > MAX_BLOCKS ? MAX_BLOCKS : maxNB));
    if (NB > rows) NB = rows;
    const int rowsPerBlock = (rows + NB - 1) / NB;

    enc_colsum_partial<<<NB, 256, 0, stream>>>(x, partial, rows, rowsPerBlock);
    enc_finish_wmma<<<1, 256, 0, stream>>>(partial, W, out, NB);
}